// SparseConvWithSkips_64561948393729
// MI455X (gfx1250) — compile-verified
//
#include <hip/hip_runtime.h>

typedef __attribute__((ext_vector_type(2))) float v2f;
typedef __attribute__((ext_vector_type(4))) float v4f;
typedef __attribute__((ext_vector_type(8))) float v8f;

#define NVOX 500000
#define KTAP 27
#define CCH  8
#define WAVES_PER_BLOCK 8
#define TILES (NVOX / 16)   // 31250, exact

// One wave32 computes a 16-voxel tile:
//   D[16x16] = sum over 54 K-chunks of A(16x4 = W^T slice) x B(4x16 = gathered feats)
// M = output channel d (0..7 valid, 8..15 zero-padded), N = voxel, K = (k,c) chunks of 4.
// K-chunk channel order is permuted to {0,1,4,5} / {2,3,6,7} so each lane's A and B
// data is one contiguous 16B load (ds_load_b128 / global_load_b128) per tap.
__global__ __launch_bounds__(256) void sparse_conv_wmma(
    const float* __restrict__ xin,      // [N][8] gather source
    const float* __restrict__ w,        // [27][8][8] layer weight slice (k,c,d)
    const int*   __restrict__ nbr,      // [N][27], -1 = missing
    float* __restrict__ out,            // [N][8]
    const float* __restrict__ residual) // non-null: out = conv + residual; null: out = relu(conv)
{
    // Weights transposed to wt[k][d][c] and zero-padded to d=0..15 so the
    // A-matrix rows 8..15 are zero without branches. 27*16*8*4 = 13824 B LDS.
    __shared__ __align__(16) float wt[KTAP * 16 * 8];
    for (int s = threadIdx.x; s < KTAP * 16 * 8; s += blockDim.x) {
        int k  = s >> 7;          // /128
        int dd = (s >> 3) & 15;
        int c  = s & 7;
        wt[s] = (dd < 8) ? w[(k * 8 + c) * 8 + dd] : 0.0f;
    }
    __syncthreads();

    const int lane = threadIdx.x & 31;
    const int wave = threadIdx.x >> 5;
    const int tile = blockIdx.x * WAVES_PER_BLOCK + wave;
    if (tile >= TILES) return;

    const int hi  = lane >> 4;       // 0: K rows 0,1   1: K rows 2,3 of each chunk
    const int dd  = lane & 15;       // A row (channel d) / B column (voxel-in-tile)
    const int vox = tile * 16 + dd;
    const int h4  = hi * 4;          // per-lane float offset for the 16B load

    // Preload the 27 neighbor indices for this lane's voxel.
    const int* nb = nbr + (size_t)vox * KTAP;
    int idxs[KTAP];
#pragma unroll
    for (int k = 0; k < KTAP; ++k) idxs[k] = nb[k];

    v8f acc00 = {}, acc01 = {};      // even taps: chunk0 / chunk1
    v8f acc10 = {}, acc11 = {};      // odd  taps: chunk0 / chunk1
#pragma unroll
    for (int k = 0; k < KTAP; ++k) {
        const int   idx = idxs[k];
        const float msk = (idx >= 0) ? 1.0f : 0.0f;          // branchless, EXEC stays full
        const float* xr = xin + (size_t)(idx < 0 ? 0 : idx) * CCH + h4;
        const float* wk = wt + k * 128 + dd * 8 + h4;

        v4f a = *(const v4f*)wk;     // one ds_load_b128: W[k][{h4..h4+3}][d]
        v4f b = *(const v4f*)xr;     // one global_load_b128: x[idx][{h4..h4+3}]
        b *= msk;

        v2f a0 = {a.x, a.y};         // chunk0: channels {0,1,4,5}
        v2f a1 = {a.z, a.w};         // chunk1: channels {2,3,6,7}
        v2f b0 = {b.x, b.y};
        v2f b1 = {b.z, b.w};

        if (k & 1) {
            acc10 = __builtin_amdgcn_wmma_f32_16x16x4_f32(false, a0, false, b0,
                                                          (short)0, acc10, false, false);
            acc11 = __builtin_amdgcn_wmma_f32_16x16x4_f32(false, a1, false, b1,
                                                          (short)0, acc11, false, false);
        } else {
            acc00 = __builtin_amdgcn_wmma_f32_16x16x4_f32(false, a0, false, b0,
                                                          (short)0, acc00, false, false);
            acc01 = __builtin_amdgcn_wmma_f32_16x16x4_f32(false, a1, false, b1,
                                                          (short)0, acc01, false, false);
        }
    }
    v8f acc = (acc00 + acc01) + (acc10 + acc11);

    // D layout: VGPR r, lanes 0..15 hold (M=r, N=lane) -> lane L owns all 8
    // channels of voxel tile*16+L as one contiguous 32B store. Lanes 16..31
    // hold the zero-padded rows M=8..15 -> skip.
    if (lane < 16) {
        float* o = out + (size_t)vox * CCH;
        if (residual) {
            const float* r = residual + (size_t)vox * CCH;
#pragma unroll
            for (int i = 0; i < 8; ++i) o[i] = acc[i] + r[i];
        } else {
#pragma unroll
            for (int i = 0; i < 8; ++i) o[i] = fmaxf(acc[i], 0.0f);
        }
    }
}

extern "C" void kernel_launch(void* const* d_in, const int* in_sizes, int n_in,
                              void* d_out, int out_size, void* d_ws, size_t ws_size,
                              hipStream_t stream) {
    const float* feats0 = (const float*)d_in[0];
    const float* feats1 = (const float*)d_in[1];
    const float* W0     = (const float*)d_in[2];   // [3][27][8][8]
    const float* W1     = (const float*)d_in[3];
    const int*   nbr0   = (const int*)d_in[4];     // [N][27]
    const int*   nbr1   = (const int*)d_in[5];
    float*       out    = (float*)d_out;           // [2N][8] concatenated

    // ping-pong intermediates in workspace (2 * 16 MB)
    float* tmpA = (float*)d_ws;
    float* tmpB = tmpA + (size_t)NVOX * CCH;

    const dim3 grid((TILES + WAVES_PER_BLOCK - 1) / WAVES_PER_BLOCK);
    const dim3 block(256);
    const size_t WL = (size_t)KTAP * CCH * CCH;    // 1728 floats per layer

    for (int g = 0; g < 2; ++g) {
        const float* x0  = g ? feats1 : feats0;
        const float* W   = g ? W1 : W0;
        const int*   nbr = g ? nbr1 : nbr0;
        float*       og  = out + (size_t)g * NVOX * CCH;

        sparse_conv_wmma<<<grid, block, 0, stream>>>(x0,   W + 0 * WL, nbr, tmpA, nullptr);
        sparse_conv_wmma<<<grid, block, 0, stream>>>(tmpA, W + 1 * WL, nbr, tmpB, nullptr);
        sparse_conv_wmma<<<grid, block, 0, stream>>>(tmpB, W + 2 * WL, nbr, og,   x0);
    }
}